// KNN_2568390443207
// MI455X (gfx1250) — compile-verified
//
#include <hip/hip_runtime.h>

typedef __attribute__((ext_vector_type(2))) float v2f;
typedef __attribute__((ext_vector_type(8))) float v8f;

#define B_DIM 2
#define N_REF 16384
#define M_QRY 8192
#define K_NN  16
#define WAVES 4
#define TILES_PER_GROUP (N_REF / 16)  // 1024 tiles of 16 refs

// One block = 4 waves on the same 16-query group; each wave owns an
// interleaved 1/4 of the ref dimension. Per 16-ref tile, one
// v_wmma_f32_16x16x4_f32 computes (exact fp32):
//   d2[r][q] = A(16x4: [rx,ry,rz,r2]) x B(4x16: [-2qx,-2qy,-2qz,1]) + C(q2[q])
// With refs on rows / queries on cols, the D layout puts 8 distances for
// lane L's own query (q = L%16) directly in lane L's registers -> top-16
// selection runs on WMMA output with zero data movement. B and C are
// loop-invariant. A final 8-way LDS merge emits sorted distances + indices.
__global__ __launch_bounds__(WAVES * 32) void knn_wmma_kernel(
    const float* __restrict__ ref, const float* __restrict__ qry,
    float* __restrict__ outD, int* __restrict__ outI)
{
    __shared__ float mD[WAVES * 32][K_NN];
    __shared__ int   mI[WAVES * 32][K_NN];

    const int tid  = threadIdx.x;
    const int wave = tid >> 5;
    const int lane = tid & 31;
    const int m    = lane & 15;   // this lane's query (B col) and ref row (A)
    const int kh   = lane >> 4;   // K-half of A/B fragments; also D row half

    const int group = blockIdx.x;                 // 0 .. B*(M/16)-1
    const int b     = group / (M_QRY / 16);
    const int m0    = (group % (M_QRY / 16)) * 16;

    // Loop-invariant query fragment: B col m = [-2qx, -2qy, -2qz, 1]
    const size_t qbase = ((size_t)b * M_QRY + (size_t)(m0 + m)) * 3;
    const float qx = qry[qbase + 0];
    const float qy = qry[qbase + 1];
    const float qz = qry[qbase + 2];
    const float q2 = qx * qx + qy * qy + qz * qz;

    v2f bfrag;
    bfrag.x = kh ? (-2.0f * qz) : (-2.0f * qx);
    bfrag.y = kh ? 1.0f         : (-2.0f * qy);

    // C: q2 broadcast along rows -> every element of this lane's C is its q2
    v8f cfrag;
#pragma unroll
    for (int i = 0; i < 8; ++i) cfrag[i] = q2;

    float best[K_NN];
    int   bid[K_NN];
#pragma unroll
    for (int i = 0; i < K_NN; ++i) { best[i] = __builtin_inff(); bid[i] = 0; }

    for (int t = wave; t < TILES_PER_GROUP; t += WAVES) {
        const int    n0    = t * 16;
        const size_t rbase = ((size_t)b * N_REF + (size_t)(n0 + m)) * 3;

        // Prefetch this wave's next ref tile into near caches
        __builtin_prefetch(&ref[rbase + (size_t)(WAVES * 16) * 3], 0, 3);

        const float rx = ref[rbase + 0];
        const float ry = ref[rbase + 1];
        const float rz = ref[rbase + 2];
        const float r2 = rx * rx + ry * ry + rz * rz;

        // A fragment (16x4 f32), row = ref point n0+m:
        // lanes 0-15 hold K={0,1} = (rx,ry); lanes 16-31 hold K={2,3} = (rz,r2)
        v2f afrag;
        afrag.x = kh ? rz : rx;
        afrag.y = kh ? r2 : ry;

        v8f d2 = __builtin_amdgcn_wmma_f32_16x16x4_f32(
            false, afrag, false, bfrag, (short)0, cfrag, false, false);

        // Lane L holds d2 for its query vs refs n0 + i + 8*kh, i = 0..7
#pragma unroll
        for (int i = 0; i < 8; ++i) {
            const float dd = d2[i];
            const int   id = n0 + i + 8 * kh;
            if (dd < best[K_NN - 1]) {
                // Branchless sorted insert: constant indices after unroll ->
                // pure v_cmp/v_cndmask, no register-array shuffles.
#pragma unroll
                for (int s = K_NN - 1; s > 0; --s) {
                    const bool  shift = dd < best[s - 1];
                    const float nb    = shift ? best[s - 1] : dd;
                    const int   ni    = shift ? bid[s - 1]  : id;
                    const bool  lt    = dd < best[s];
                    best[s] = lt ? nb : best[s];
                    bid[s]  = lt ? ni : bid[s];
                }
                const bool lt0 = dd < best[0];
                bid[0]  = lt0 ? id : bid[0];
                best[0] = lt0 ? dd : best[0];
            }
        }
    }

    // Publish per-lane sorted lists for the cross-wave merge
#pragma unroll
    for (int i = 0; i < K_NN; ++i) { mD[tid][i] = best[i]; mI[tid][i] = bid[i]; }
    __syncthreads();

    // Final merge: thread q (<16) merges its query's 8 sorted lists
    if (tid < 16) {
        int heads[2 * WAVES];
#pragma unroll
        for (int l = 0; l < 2 * WAVES; ++l) heads[l] = 0;

        const size_t obase = ((size_t)b * M_QRY + (size_t)(m0 + tid)) * K_NN;
        for (int j = 0; j < K_NN; ++j) {
            float bd = __builtin_inff();
            int   bi = 0x7fffffff;
            int   bl = 0;
#pragma unroll
            for (int l = 0; l < 2 * WAVES; ++l) {
                const int w   = l >> 1;
                const int h   = l & 1;
                const int src = w * 32 + h * 16 + tid;
                const float v  = mD[src][heads[l]];
                const int   iv = mI[src][heads[l]];
                if (v < bd || (v == bd && iv < bi)) { bd = v; bi = iv; bl = l; }
            }
            heads[bl]++;
            outD[obase + j] = sqrtf(fmaxf(bd, 0.0f));
            outI[obase + j] = bi;
        }
    }
}

extern "C" void kernel_launch(void* const* d_in, const int* in_sizes, int n_in,
                              void* d_out, int out_size, void* d_ws, size_t ws_size,
                              hipStream_t stream) {
    (void)in_sizes; (void)n_in; (void)out_size; (void)d_ws; (void)ws_size;
    const float* ref = (const float*)d_in[0];
    const float* qry = (const float*)d_in[1];
    // d_in[2] is k (=16), fixed at compile time as K_NN.

    float* outD = (float*)d_out;                                         // [B, M, 16] f32
    int*   outI = (int*)((float*)d_out + (size_t)B_DIM * M_QRY * K_NN);  // [B, M, 16] i32

    dim3 grid(B_DIM * (M_QRY / 16));  // 1024 blocks
    dim3 block(WAVES * 32);           // 128 threads = 4 waves
    knn_wmma_kernel<<<grid, block, 0, stream>>>(ref, qry, outD, outI);
}